// GCN_28252294873367
// MI455X (gfx1250) — compile-verified
//
#include <hip/hip_runtime.h>
#include <hip/hip_bf16.h>

typedef __attribute__((ext_vector_type(2))) float v2f;
typedef __attribute__((ext_vector_type(8))) float v8f;

#define EMB 300
#define FEAT 256
#define NGRAPH 2048
#define BN_EPS 1e-5f

// ---------------------------------------------------------------------------
// WMMA f32 GEMM: C[M][Ncols] = A[M][K] @ B[K][Ncols] (+bias) (+relu)
// V_WMMA_F32_16X16X4_F32, K step 4. Each wave computes a 16(M) x 64(N) strip
// (4 WMMA tiles) from a single A fragment -> 4 WMMAs per 8-byte A load.
// Block = 8 waves = 8 M-tiles sharing one 64-column B panel staged in LDS.
// B panel is stored k-pair interleaved: B[k][j] -> sB[(k>>1)*128 + 2*j + (k&1)]
// so each WMMA B fragment is one contiguous 8-byte ds_load_b64 (no shuffles).
// Requires M % 16 == 0 and K % 4 == 0 (true for all calls here).
// ---------------------------------------------------------------------------
__global__ __launch_bounds__(256)
void gemm_wmma_f32(const float* __restrict__ A, const float* __restrict__ B,
                   const float* __restrict__ bias, float* __restrict__ C,
                   int M, int K, int Ncols, int relu)
{
    __shared__ float sB[EMB * 64];            // up to 300x64 panel = 76.8 KB
    const int tid = threadIdx.x;
    const int colBase = blockIdx.y * 64;

    // Stage B[:, colBase:colBase+64) into LDS, k-pair interleaved, zero-padded.
    for (int idx = tid; idx < K * 64; idx += blockDim.x) {
        int k = idx >> 6;
        int j = idx & 63;
        int c = colBase + j;
        float v = (c < Ncols) ? B[(size_t)k * Ncols + c] : 0.0f;
        sB[(k >> 1) * 128 + 2 * j + (k & 1)] = v;
    }
    __syncthreads();

    const int wave = tid >> 5;
    const int lane = tid & 31;
    const int lrow = lane & 15;    // row within A tile / col within B,D tile
    const int hi   = lane >> 4;    // lane-group selects K pair / M+8 half
    const int tm   = blockIdx.x * 8 + wave;
    if (tm * 16 >= M) return;      // wave-uniform: EXEC stays all-ones for WMMA

    const float* Arow = A + (size_t)(tm * 16 + lrow) * K;
    v8f acc0 = {}, acc1 = {}, acc2 = {}, acc3 = {};
    for (int kk = 0; kk < K; kk += 4) {
        const int ka = kk + 2 * hi;           // even; A/B k = vgpr + 2*laneHi
        v2f a = *(const v2f*)(Arow + ka);      // 8B aligned (K and ka even)
        const float* bp = &sB[(ka >> 1) * 128 + 2 * lrow];
        v2f b;
        b = *(const v2f*)(bp + 0);
        acc0 = __builtin_amdgcn_wmma_f32_16x16x4_f32(false, a, false, b, (short)0, acc0, false, false);
        b = *(const v2f*)(bp + 32);
        acc1 = __builtin_amdgcn_wmma_f32_16x16x4_f32(false, a, false, b, (short)0, acc1, false, false);
        b = *(const v2f*)(bp + 64);
        acc2 = __builtin_amdgcn_wmma_f32_16x16x4_f32(false, a, false, b, (short)0, acc2, false, false);
        b = *(const v2f*)(bp + 96);
        acc3 = __builtin_amdgcn_wmma_f32_16x16x4_f32(false, a, false, b, (short)0, acc3, false, false);
    }

    v8f accs[4] = {acc0, acc1, acc2, acc3};
    #pragma unroll
    for (int t = 0; t < 4; ++t) {
        int col = colBase + t * 16 + lrow;
        if (col < Ncols) {
            const float bv = bias ? bias[col] : 0.0f;
            #pragma unroll
            for (int v = 0; v < 8; ++v) {     // D vgpr v -> rows v (lo) / v+8 (hi)
                int row = tm * 16 + v + 8 * hi;
                float r = accs[t][v] + bv;
                if (relu) r = fmaxf(r, 0.0f);
                C[(size_t)row * Ncols + col] = r;
            }
        }
    }
}

// ---------------------------------------------------------------------------
// h[n][c] = x_emb1[x[n,0]][c] + x_emb2[x[n,1]][c]
// ---------------------------------------------------------------------------
__global__ void embed_kernel(const int* __restrict__ x,
                             const float* __restrict__ e1,
                             const float* __restrict__ e2,
                             float* __restrict__ h, int N)
{
    int total = N * EMB;
    for (int i = blockIdx.x * blockDim.x + threadIdx.x; i < total;
         i += gridDim.x * blockDim.x) {
        int n = i / EMB;
        int c = i - n * EMB;
        h[i] = e1[x[2 * n] * EMB + c] + e2[x[2 * n + 1] * EMB + c];
    }
}

// agg[n][c] = conv_b[c]   (seed with bias so scatter produces segsum + b)
__global__ void init_agg_kernel(float* __restrict__ agg,
                                const float* __restrict__ b, int N)
{
    int total = N * EMB;
    for (int i = blockIdx.x * blockDim.x + threadIdx.x; i < total;
         i += gridDim.x * blockDim.x) {
        int c = i % EMB;
        agg[i] = b[c];
    }
}

__global__ void zero_kernel(float* __restrict__ p, int n)
{
    for (int i = blockIdx.x * blockDim.x + threadIdx.x; i < n;
         i += gridDim.x * blockDim.x) p[i] = 0.0f;
}

// ---------------------------------------------------------------------------
// agg[dst[e]][c] += hw[src[e]][c] + escalar(e)   over E real edges + N self loops
// (hw is 120 MB -> resident in the 192 MB L2; atomics run at L2 rate)
// ---------------------------------------------------------------------------
__global__ void edge_agg_kernel(const float* __restrict__ hw,
                                const int* __restrict__ src,
                                const int* __restrict__ dst,
                                const int* __restrict__ eattr,
                                const float* __restrict__ ee1, // [6]
                                const float* __restrict__ ee2, // [3]
                                float* __restrict__ agg, int E, int N)
{
    int total = (E + N) * EMB;
    for (int i = blockIdx.x * blockDim.x + threadIdx.x; i < total;
         i += gridDim.x * blockDim.x) {
        int e = i / EMB;
        int c = i - e * EMB;
        int s, d;
        float ev;
        if (e < E) {
            s = src[e];
            d = dst[e];
            ev = ee1[eattr[2 * e]] + ee2[eattr[2 * e + 1]];
        } else {                               // self loop: bond type 4, dir 0
            s = d = e - E;
            ev = ee1[4] + ee2[0];
        }
        atomicAdd(&agg[(size_t)d * EMB + c], hw[(size_t)s * EMB + c] + ev);
    }
}

// ---------------------------------------------------------------------------
// BatchNorm stats: per-channel sum and sum-of-squares (LDS atomics -> global)
// ---------------------------------------------------------------------------
__global__ __launch_bounds__(256)
void bn_stats_kernel(const float* __restrict__ agg,
                     float* __restrict__ gsum, float* __restrict__ gsq, int N)
{
    __shared__ float ssum[EMB];
    __shared__ float ssq[EMB];
    for (int c = threadIdx.x; c < EMB; c += blockDim.x) { ssum[c] = 0.f; ssq[c] = 0.f; }
    __syncthreads();
    int total = N * EMB;
    for (int i = blockIdx.x * blockDim.x + threadIdx.x; i < total;
         i += gridDim.x * blockDim.x) {
        int c = i % EMB;
        float v = agg[i];
        atomicAdd(&ssum[c], v);
        atomicAdd(&ssq[c], v * v);
    }
    __syncthreads();
    for (int c = threadIdx.x; c < EMB; c += blockDim.x) {
        atomicAdd(&gsum[c], ssum[c]);
        atomicAdd(&gsq[c], ssq[c]);
    }
}

__global__ void bn_apply_kernel(const float* __restrict__ agg,
                                const float* __restrict__ gsum,
                                const float* __restrict__ gsq,
                                const float* __restrict__ gamma,
                                const float* __restrict__ beta,
                                float* __restrict__ h, int N)
{
    float invN = 1.0f / (float)N;
    int total = N * EMB;
    for (int i = blockIdx.x * blockDim.x + threadIdx.x; i < total;
         i += gridDim.x * blockDim.x) {
        int c = i % EMB;
        float mean = gsum[c] * invN;
        float var  = gsq[c] * invN - mean * mean;   // biased, matches reference
        float r = (agg[i] - mean) * rsqrtf(var + BN_EPS) * gamma[c] + beta[c];
        h[i] = fmaxf(r, 0.0f);
    }
}

// ---------------------------------------------------------------------------
// Global mean pool: sums + counts via atomics, then divide in place.
// ---------------------------------------------------------------------------
__global__ void pool_kernel(const float* __restrict__ hf,
                            const int* __restrict__ batch,
                            float* __restrict__ g, float* __restrict__ counts, int N)
{
    int total = N * FEAT;
    for (int i = blockIdx.x * blockDim.x + threadIdx.x; i < total;
         i += gridDim.x * blockDim.x) {
        int n = i >> 8;              // /256
        int c = i & 255;
        int b = batch[n];
        atomicAdd(&g[(size_t)b * FEAT + c], hf[i]);
        if (c == 0) atomicAdd(&counts[b], 1.0f);
    }
}

__global__ void pool_div_kernel(float* __restrict__ g,
                                const float* __restrict__ counts)
{
    int total = NGRAPH * FEAT;
    for (int i = blockIdx.x * blockDim.x + threadIdx.x; i < total;
         i += gridDim.x * blockDim.x) {
        g[i] /= fmaxf(counts[i >> 8], 1.0f);
    }
}

// ---------------------------------------------------------------------------
extern "C" void kernel_launch(void* const* d_in, const int* in_sizes, int n_in,
                              void* d_out, int out_size, void* d_ws, size_t ws_size,
                              hipStream_t stream)
{
    const int*   x          = (const int*)d_in[0];
    const int*   edge_index = (const int*)d_in[1];
    const int*   edge_attr  = (const int*)d_in[2];
    const int*   batch      = (const int*)d_in[3];
    const float* x_emb1     = (const float*)d_in[4];
    const float* x_emb2     = (const float*)d_in[5];
    const float* conv_w     = (const float*)d_in[6];
    const float* conv_b     = (const float*)d_in[7];
    const float* eemb1      = (const float*)d_in[8];
    const float* eemb2      = (const float*)d_in[9];
    const float* bn_gamma   = (const float*)d_in[10];
    const float* bn_beta    = (const float*)d_in[11];
    const float* feat_w     = (const float*)d_in[12];
    const float* feat_b     = (const float*)d_in[13];
    const float* pred_w1    = (const float*)d_in[14];
    const float* pred_b1    = (const float*)d_in[15];
    const float* pred_w2    = (const float*)d_in[16];
    const float* pred_b2    = (const float*)d_in[17];

    const int N = in_sizes[0] / 2;      // 100000
    const int E = in_sizes[1] / 2;      // 200000
    const int* src = edge_index;        // row 0
    const int* dst = edge_index + E;    // row 1

    // ---- workspace carve-up (ping-ponged; ~360 MB total) ----
    char* wsp = (char*)d_ws;
    size_t off = 0;
    auto carve = [&](size_t bytes) {
        void* p = wsp + off;
        off += (bytes + 255) & ~(size_t)255;
        return p;
    };
    float* h     = (float*)carve((size_t)N * EMB * 4);
    float* hw    = (float*)carve((size_t)N * EMB * 4);  // reused as hfeat (N*256)
    float* agg   = (float*)carve((size_t)N * EMB * 4);  // reused as g2 (2048*128)
    float* bnsum = (float*)carve(EMB * 4);
    float* bnsq  = (float*)carve(EMB * 4);
    float* counts= (float*)carve(NGRAPH * 4);
    float* gpool = (float*)d_out;                  // [2048][256]
    float* pred  = (float*)d_out + NGRAPH * FEAT;  // [2048][2]
    float* hfeat = hw;
    float* g2    = agg;
    (void)ws_size; (void)n_in; (void)out_size;

    const int TPB = 256;
    auto nblk = [&](int total) { return (total + TPB - 1) / TPB; };
    auto gemm = [&](const float* A, const float* B, const float* bias, float* C,
                    int M, int K, int Ncols, int relu) {
        dim3 grid((M / 16 + 7) / 8, (Ncols + 63) / 64);
        gemm_wmma_f32<<<grid, 256, 0, stream>>>(A, B, bias, C, M, K, Ncols, relu);
    };

    // node embedding
    embed_kernel<<<nblk(N * EMB), TPB, 0, stream>>>(x, x_emb1, x_emb2, h, N);

    // 5 GIN conv layers
    for (int l = 0; l < 5; ++l) {
        gemm(h, conv_w + (size_t)l * EMB * EMB, nullptr, hw, N, EMB, EMB, 0);
        init_agg_kernel<<<nblk(N * EMB), TPB, 0, stream>>>(agg, conv_b + l * EMB, N);
        edge_agg_kernel<<<nblk((E + N) * EMB), TPB, 0, stream>>>(
            hw, src, dst, edge_attr, eemb1 + l * 6, eemb2 + l * 3, agg, E, N);
        zero_kernel<<<1, 512, 0, stream>>>(bnsum, EMB);
        zero_kernel<<<1, 512, 0, stream>>>(bnsq, EMB);
        bn_stats_kernel<<<1024, 256, 0, stream>>>(agg, bnsum, bnsq, N);
        bn_apply_kernel<<<nblk(N * EMB), TPB, 0, stream>>>(
            agg, bnsum, bnsq, bn_gamma + l * EMB, bn_beta + l * EMB, h, N);
    }

    // feature projection 300 -> 256
    gemm(h, feat_w, feat_b, hfeat, N, EMB, FEAT, 0);

    // global mean pool
    zero_kernel<<<nblk(NGRAPH * FEAT), TPB, 0, stream>>>(gpool, NGRAPH * FEAT);
    zero_kernel<<<1, 512, 0, stream>>>(counts, NGRAPH);
    pool_kernel<<<nblk(N * FEAT), TPB, 0, stream>>>(hfeat, batch, gpool, counts, N);
    pool_div_kernel<<<nblk(NGRAPH * FEAT), TPB, 0, stream>>>(gpool, counts);

    // prediction MLP: relu(g @ w1 + b1) @ w2 + b2
    gemm(gpool, pred_w1, pred_b1, g2, NGRAPH, FEAT, FEAT / 2, 1);
    gemm(g2, pred_w2, pred_b2, pred, NGRAPH, FEAT / 2, 2, 0);
}